// JKNet_65111704207521
// MI455X (gfx1250) — compile-verified
//
#include <hip/hip_runtime.h>
#include <hip/hip_bf16.h>

typedef __attribute__((ext_vector_type(2))) float v2f;
typedef __attribute__((ext_vector_type(8))) float v8f;

#define HIDW 64

// ---------------------------------------------------------------------------
// Elementwise helpers
// ---------------------------------------------------------------------------
__global__ void fill_kernel(float* __restrict__ p, float v, int n) {
    int i = blockIdx.x * blockDim.x + threadIdx.x;
    if (i < n) p[i] = v;
}

__global__ void degree_kernel(const int* __restrict__ src, const int* __restrict__ dst,
                              float* __restrict__ deg_s, float* __restrict__ deg_d, int E) {
    int i = blockIdx.x * blockDim.x + threadIdx.x;
    if (i < E) {
        atomicAdd(&deg_s[src[i]], 1.0f);
        atomicAdd(&deg_d[dst[i]], 1.0f);
    }
}

__global__ void invnorm_kernel(float* __restrict__ p, int n) {
    int i = blockIdx.x * blockDim.x + threadIdx.x;
    if (i < n) {
        float v = p[i];
        v = fmaxf(v, 1.0f);
        p[i] = 1.0f / sqrtf(v);
    }
}

// agg -> relu(agg * norm_d[n] + b[c]), in place on [N,64]
__global__ void bias_relu_kernel(float* __restrict__ H, const float* __restrict__ norm_d,
                                 const float* __restrict__ b, int N) {
    int i = blockIdx.x * blockDim.x + threadIdx.x;
    if (i < N * HIDW) {
        int n = i >> 6;
        int c = i & 63;
        float v = H[i] * norm_d[n] + b[c];
        H[i] = v > 0.0f ? v : 0.0f;
    }
}

__global__ void outinit_kernel(float* __restrict__ out, const float* __restrict__ bout, int N) {
    int i = blockIdx.x * blockDim.x + threadIdx.x;
    if (i < N * HIDW) out[i] = bout[i & 63];
}

// ---------------------------------------------------------------------------
// Edge scatter: out[dst[e]][c] += Hsrc[src[e]][c], 64 columns per edge.
// Thread = (edge, col); reads coalesce across the 64-lane span of one edge,
// fp32 global atomics land in L2 (working set << 192MB L2).
// ---------------------------------------------------------------------------
__global__ void scatter64_kernel(const float* __restrict__ Hsrc,
                                 const int* __restrict__ src, const int* __restrict__ dst,
                                 float* __restrict__ out, int E) {
    long long i = (long long)blockIdx.x * blockDim.x + threadIdx.x;
    long long tot = (long long)E * HIDW;
    if (i < tot) {
        int e = (int)(i >> 6);
        int c = (int)(i & 63);
        float v = Hsrc[(size_t)src[e] * HIDW + c];
        atomicAdd(&out[(size_t)dst[e] * HIDW + c], v);
    }
}

// ---------------------------------------------------------------------------
// WMMA fp32 GEMM: Out[m][n] = sum_k Asel(m,k) * W[k][n]   (n in [0,64))
//   A column k <  ksplit  comes from A1 (row stride lda1)
//   A column k >= ksplit  comes from A2 (row stride lda2, col k-ksplit)
//   optional per-row scale (norm) applied to A.
// One wave computes a 16x64 output slab using 4x V_WMMA_F32_16X16X4_F32
// accumulators, stepping K by 4.
// A 16x4 f32 fragment: lane(half h, lm): elem0=A[row0+lm][k0+2h], elem1=+1
// B 4x16  fragment: lane: elem0=W[k0+2h][n0+lm], elem1=W[k0+2h+1][n0+lm]
// C/D 16x16: vgpr r holds row row0+r+8h, col n0+lm
// ---------------------------------------------------------------------------
__global__ __launch_bounds__(256) void gemm_wmma_k(
    const float* __restrict__ A1, int lda1,
    const float* __restrict__ A2, int lda2,
    int ksplit, int K,
    const float* __restrict__ W,      // K x 64 row-major
    const float* __restrict__ norm,   // may be null
    float* __restrict__ Out,          // nrows x 64
    int nrows) {
    int wave = (blockIdx.x * blockDim.x + threadIdx.x) >> 5;
    int row0 = wave * 16;
    if (row0 >= nrows) return;   // wave-uniform: EXEC stays all-1s for WMMA

    int lane = threadIdx.x & 31;
    int half = lane >> 4;        // 0: K pair {k0,k0+1} / rows M..M+7 ; 1: {k0+2,k0+3} / M+8..
    int lm   = lane & 15;
    int arow = row0 + lm;
    int arowc = arow < nrows ? arow : nrows - 1;
    float scale = (arow < nrows) ? (norm ? norm[arowc] : 1.0f) : 0.0f;

    v8f acc[4] = {};

    for (int k0 = 0; k0 < K; k0 += 4) {
        int ka = k0 + 2 * half;               // ksplit % 4 == 0 -> wave-uniform branch
        const float* Ap;
        int kk;
        if (ka < ksplit) { Ap = A1 + (size_t)arowc * lda1; kk = ka; }
        else             { Ap = A2 + (size_t)arowc * lda2; kk = ka - ksplit; }
        v2f a = *(const v2f*)(Ap + kk);       // 8B-aligned (kk even, row base 16B-aligned)
        a *= scale;

        const float* Wk  = W + (size_t)ka * HIDW;
        const float* Wk1 = Wk + HIDW;
#pragma unroll
        for (int t = 0; t < 4; ++t) {
            int n = t * 16 + lm;
            v2f b;
            b.x = Wk[n];
            b.y = Wk1[n];
            acc[t] = __builtin_amdgcn_wmma_f32_16x16x4_f32(
                /*neg_a=*/false, a, /*neg_b=*/false, b,
                /*c_mod=*/(short)0, acc[t], /*reuse_a=*/false, /*reuse_b=*/false);
        }
    }

#pragma unroll
    for (int t = 0; t < 4; ++t) {
#pragma unroll
        for (int r = 0; r < 8; ++r) {
            int m = row0 + r + 8 * half;
            if (m < nrows) Out[(size_t)m * HIDW + t * 16 + lm] = acc[t][r];
        }
    }
}

// ---------------------------------------------------------------------------
// Orchestration
// Inputs: feats[N,128], src[E], dst[E], W1[128,64], b1[64], W2[64,64], b2[64],
//         Wout[128,64], bout[64].  Output: [N,64] fp32.
// Key trick: segment_sum(J[src]) @ Wout == segment_sum((J @ Wout)[src]) —
// project before the JK scatter, halving the final scatter payload.
// ---------------------------------------------------------------------------
extern "C" void kernel_launch(void* const* d_in, const int* in_sizes, int n_in,
                              void* d_out, int out_size, void* d_ws, size_t ws_size,
                              hipStream_t stream) {
    const float* feats = (const float*)d_in[0];
    const int*   src   = (const int*)d_in[1];
    const int*   dst   = (const int*)d_in[2];
    const float* W1    = (const float*)d_in[3];
    const float* b1    = (const float*)d_in[4];
    const float* W2    = (const float*)d_in[5];
    const float* b2    = (const float*)d_in[6];
    const float* Wout  = (const float*)d_in[7];
    const float* bout  = (const float*)d_in[8];

    const int IN_DIM = 128;
    int N = in_sizes[0] / IN_DIM;
    int E = in_sizes[1];
    float* out = (float*)d_out;

    float* ws     = (float*)d_ws;
    float* norm_s = ws;                         // N
    float* norm_d = ws + N;                     // N
    float* H      = ws + 2 * (size_t)N;         // N*64  (GEMM output / projection P)
    float* H1     = H  + (size_t)N * HIDW;      // N*64
    float* H2     = H1 + (size_t)N * HIDW;      // N*64

    const int T = 256;
    int nh = N * HIDW;
    int gN2   = (2 * N + T - 1) / T;
    int gE    = (E + T - 1) / T;
    int gNH   = (nh + T - 1) / T;
    long long tot = (long long)E * HIDW;
    int gScat = (int)((tot + T - 1) / T);
    int waves  = (N + 15) / 16;
    int gGemm  = (waves + 7) / 8;               // 8 waves (256 thr) per block

    // degrees -> symmetric norms
    fill_kernel<<<gN2, T, 0, stream>>>(norm_s, 0.0f, 2 * N);
    degree_kernel<<<gE, T, 0, stream>>>(src, dst, norm_s, norm_d, E);
    invnorm_kernel<<<gN2, T, 0, stream>>>(norm_s, 2 * N);

    // layer 1: H = (X . norm_s) @ W1 ; h1 = relu(scatter(H)*norm_d + b1)
    gemm_wmma_k<<<gGemm, T, 0, stream>>>(feats, IN_DIM, feats, IN_DIM, IN_DIM, IN_DIM,
                                         W1, norm_s, H, N);
    fill_kernel<<<gNH, T, 0, stream>>>(H1, 0.0f, nh);
    scatter64_kernel<<<gScat, T, 0, stream>>>(H, src, dst, H1, E);
    bias_relu_kernel<<<gNH, T, 0, stream>>>(H1, norm_d, b1, N);

    // layer 2: H = (h1 . norm_s) @ W2 ; h2 = relu(scatter(H)*norm_d + b2)
    gemm_wmma_k<<<gGemm, T, 0, stream>>>(H1, HIDW, H1, HIDW, HIDW, HIDW,
                                         W2, norm_s, H, N);
    fill_kernel<<<gNH, T, 0, stream>>>(H2, 0.0f, nh);
    scatter64_kernel<<<gScat, T, 0, stream>>>(H, src, dst, H2, E);
    bias_relu_kernel<<<gNH, T, 0, stream>>>(H2, norm_d, b2, N);

    // JK projection-first: P = h1 @ Wout[0:64,:] + h2 @ Wout[64:128,:]  (into H)
    gemm_wmma_k<<<gGemm, T, 0, stream>>>(H1, HIDW, H2, HIDW, HIDW, 2 * HIDW,
                                         Wout, nullptr, H, N);

    // out = bout + scatter(P)
    outinit_kernel<<<gNH, T, 0, stream>>>(out, bout, N);
    scatter64_kernel<<<gScat, T, 0, stream>>>(H, src, dst, out, E);
}